// InstanceSimilarityModelClassic_40243843563530
// MI455X (gfx1250) — compile-verified
//
#include <hip/hip_runtime.h>
#include <math.h>

// Volume shape (D,H,W) and center of gravity from the reference.
constexpr int DD = 192;
constexpr int HH = 224;
constexpr int WW = 192;

typedef float v2f __attribute__((ext_vector_type(2)));
typedef float v8f __attribute__((ext_vector_type(8)));

// ---------------------------------------------------------------------------
// Kernel 1: compose the 4x4 transform T = inv(flo) @ T_rig @ ref  (one wave).
// T_rig = translate(cog+t) @ [R*diag(s) | R@(-s*cog)] (row 3 = 0,0,0,1)
// ---------------------------------------------------------------------------
__device__ inline void mm3(float* o, const float* a, const float* b) {
    for (int i = 0; i < 3; ++i)
        for (int j = 0; j < 3; ++j)
            o[i * 3 + j] = a[i * 3 + 0] * b[0 * 3 + j] +
                           a[i * 3 + 1] * b[1 * 3 + j] +
                           a[i * 3 + 2] * b[2 * 3 + j];
}

__device__ inline void mm4(float* o, const float* a, const float* b) {
    for (int i = 0; i < 4; ++i)
        for (int j = 0; j < 4; ++j)
            o[i * 4 + j] = a[i * 4 + 0] * b[0 * 4 + j] +
                           a[i * 4 + 1] * b[1 * 4 + j] +
                           a[i * 4 + 2] * b[2 * 4 + j] +
                           a[i * 4 + 3] * b[3 * 4 + j];
}

// General 4x4 inverse via adjugate (row-major in == row-major out).
__device__ inline void inv4(const float* m, float* inv) {
    inv[0]  =  m[5]*m[10]*m[15] - m[5]*m[11]*m[14] - m[9]*m[6]*m[15] + m[9]*m[7]*m[14] + m[13]*m[6]*m[11] - m[13]*m[7]*m[10];
    inv[4]  = -m[4]*m[10]*m[15] + m[4]*m[11]*m[14] + m[8]*m[6]*m[15] - m[8]*m[7]*m[14] - m[12]*m[6]*m[11] + m[12]*m[7]*m[10];
    inv[8]  =  m[4]*m[9]*m[15]  - m[4]*m[11]*m[13] - m[8]*m[5]*m[15] + m[8]*m[7]*m[13] + m[12]*m[5]*m[11] - m[12]*m[7]*m[9];
    inv[12] = -m[4]*m[9]*m[14]  + m[4]*m[10]*m[13] + m[8]*m[5]*m[14] - m[8]*m[6]*m[13] - m[12]*m[5]*m[10] + m[12]*m[6]*m[9];
    inv[1]  = -m[1]*m[10]*m[15] + m[1]*m[11]*m[14] + m[9]*m[2]*m[15] - m[9]*m[3]*m[14] - m[13]*m[2]*m[11] + m[13]*m[3]*m[10];
    inv[5]  =  m[0]*m[10]*m[15] - m[0]*m[11]*m[14] - m[8]*m[2]*m[15] + m[8]*m[3]*m[14] + m[12]*m[2]*m[11] - m[12]*m[3]*m[10];
    inv[9]  = -m[0]*m[9]*m[15]  + m[0]*m[11]*m[13] + m[8]*m[1]*m[15] - m[8]*m[3]*m[13] - m[12]*m[1]*m[11] + m[12]*m[3]*m[9];
    inv[13] =  m[0]*m[9]*m[14]  - m[0]*m[10]*m[13] - m[8]*m[1]*m[14] + m[8]*m[2]*m[13] + m[12]*m[1]*m[10] - m[12]*m[2]*m[9];
    inv[2]  =  m[1]*m[6]*m[15]  - m[1]*m[7]*m[14]  - m[5]*m[2]*m[15] + m[5]*m[3]*m[14] + m[13]*m[2]*m[7]  - m[13]*m[3]*m[6];
    inv[6]  = -m[0]*m[6]*m[15]  + m[0]*m[7]*m[14]  + m[4]*m[2]*m[15] - m[4]*m[3]*m[14] - m[12]*m[2]*m[7]  + m[12]*m[3]*m[6];
    inv[10] =  m[0]*m[5]*m[15]  - m[0]*m[7]*m[13]  - m[4]*m[1]*m[15] + m[4]*m[3]*m[13] + m[12]*m[1]*m[7]  - m[12]*m[3]*m[5];
    inv[14] = -m[0]*m[5]*m[14]  + m[0]*m[6]*m[13]  + m[4]*m[1]*m[14] - m[4]*m[2]*m[13] - m[12]*m[1]*m[6]  + m[12]*m[2]*m[5];
    inv[3]  = -m[1]*m[6]*m[11]  + m[1]*m[7]*m[10]  + m[5]*m[2]*m[11] - m[5]*m[3]*m[10] - m[9]*m[2]*m[7]   + m[9]*m[3]*m[6];
    inv[7]  =  m[0]*m[6]*m[11]  - m[0]*m[7]*m[10]  - m[4]*m[2]*m[11] + m[4]*m[3]*m[10] + m[8]*m[2]*m[7]   - m[8]*m[3]*m[6];
    inv[11] = -m[0]*m[5]*m[11]  + m[0]*m[7]*m[9]   + m[4]*m[1]*m[11] - m[4]*m[3]*m[9]  - m[8]*m[1]*m[7]   + m[8]*m[3]*m[5];
    inv[15] =  m[0]*m[5]*m[10]  - m[0]*m[6]*m[9]   - m[4]*m[1]*m[10] + m[4]*m[2]*m[9]  + m[8]*m[1]*m[6]   - m[8]*m[2]*m[5];
    float det = m[0]*inv[0] + m[1]*inv[4] + m[2]*inv[8] + m[3]*inv[12];
    det = 1.0f / det;
    for (int i = 0; i < 16; ++i) inv[i] *= det;
}

__global__ void compose_T_kernel(const float* __restrict__ ang,
                                 const float* __restrict__ tr,
                                 const float* __restrict__ sc,
                                 const float* __restrict__ ref,
                                 const float* __restrict__ flo,
                                 float* __restrict__ Tout) {
    if (threadIdx.x != 0 || blockIdx.x != 0) return;

    const float cx = cosf(ang[0]), sx = sinf(ang[0]);
    const float cy = cosf(ang[1]), sy = sinf(ang[1]);
    const float cz = cosf(ang[2]), sz = sinf(ang[2]);
    const float Rx[9] = {1, 0, 0,   0, cx, -sx,  0, sx, cx};
    const float Ry[9] = {cy, 0, sy, 0, 1, 0,    -sy, 0, cy};
    const float Rz[9] = {cz, -sz, 0, sz, cz, 0,  0, 0, 1};
    float Ryz[9], R[9];
    mm3(Ryz, Ry, Rz);
    mm3(R, Rx, Ryz);     // R = Rx @ Ry @ Rz

    const float s[3]   = {expf(sc[0]), expf(sc[1]), expf(sc[2])};
    const float cog[3] = {96.0f, 112.0f, 96.0f};

    // T_rig = translate(cog + t) @ rot @ scale @ translate(-cog)
    float Trig[16] = {0};
    Trig[15] = 1.0f;
    for (int i = 0; i < 3; ++i) {
        for (int j = 0; j < 3; ++j) Trig[i * 4 + j] = R[i * 3 + j] * s[j];
        Trig[i * 4 + 3] = cog[i] + tr[i]
                        - (R[i * 3 + 0] * s[0] * cog[0] +
                           R[i * 3 + 1] * s[1] * cog[1] +
                           R[i * 3 + 2] * s[2] * cog[2]);
    }

    float Finv[16], Tmp[16], T[16];
    inv4(flo, Finv);
    mm4(Tmp, Trig, ref);   // T_rig @ ref_v2r
    mm4(T, Finv, Tmp);     // inv(flo_v2r) @ ...
    for (int i = 0; i < 16; ++i) Tout[i] = T[i];
}

// ---------------------------------------------------------------------------
// Kernel 2: resample. One wave = 32 consecutive k (W is 6*32, so EXEC is
// always all-ones as WMMA requires). Coordinates for 16 voxels per
// v_wmma_f32_16x16x4_f32: A = T padded to 16x4, B = 16 homogeneous columns.
// ISA 7.12.2 layouts: A/B split K across half-waves (v0={K0|K2}, v1={K1|K3});
// D row r lives in VGPR r, column n in lane n. Lanes 16..31 pick their coords
// out of the second WMMA via one lane-permute per component.
// ---------------------------------------------------------------------------
__global__ __launch_bounds__(256)
void resample_affine_kernel(const float* __restrict__ vol,
                            const float* __restrict__ Tg,
                            float* __restrict__ out) {
    __shared__ float sT[16];
    if (threadIdx.x < 16) sT[threadIdx.x] = Tg[threadIdx.x];
    __syncthreads();

    const int lane     = threadIdx.x & 31;
    const int warp     = threadIdx.x >> 5;
    const int flatWave = blockIdx.x * (256 / 32) + warp;   // 258048 total
    const int chunk    = flatWave % (WW / 32);             // k-chunk 0..5
    int rem            = flatWave / (WW / 32);
    const int jj       = rem % HH;
    const int ii       = rem / HH;
    const int k0       = chunk * 32;

    const int m  = lane & 15;
    const int hi = lane >> 4;

    // A operand: rows of T (rows >= 3 are zero padding). Branchless + in-bounds.
    const int mr = (m < 3) ? m : 0;
    float ax = hi ? sT[4 * mr + 2] : sT[4 * mr + 0];
    float ay = hi ? sT[4 * mr + 3] : sT[4 * mr + 1];
    ax = (m < 3) ? ax : 0.0f;
    ay = (m < 3) ? ay : 0.0f;
    v2f a;  a.x = ax;  a.y = ay;

    // B operands: column n = voxel (ii, jj, k0+n, 1); lanes 0-15 carry rows
    // K0/K1 (= i, j), lanes 16-31 carry rows K2/K3 (= k, 1).
    v2f b1, b2;
    b1.x = hi ? (float)(k0 + m)      : (float)ii;
    b1.y = hi ? 1.0f                 : (float)jj;
    b2.x = hi ? (float)(k0 + 16 + m) : (float)ii;
    b2.y = b1.y;

    v8f c0 = {};
    // (neg_a, A, neg_b, B, c_mod, C, reuse_a, reuse_b)
    v8f d1 = __builtin_amdgcn_wmma_f32_16x16x4_f32(
        false, a, false, b1, (short)0, c0, false, false);
    v8f d2 = __builtin_amdgcn_wmma_f32_16x16x4_f32(
        false, a, false, b2, (short)0, c0, /*reuse_a hint illegal here*/ false, false);

    // Rows 0..2 of D = di, dj, dk for column = lane (voxels k0..k0+15 in d1,
    // k0+16..k0+31 in d2). Executed by all lanes, then selected.
    const float s0 = __shfl(d2[0], m, 32);
    const float s1 = __shfl(d2[1], m, 32);
    const float s2 = __shfl(d2[2], m, 32);
    const float di = hi ? s0 : d1[0];
    const float dj = hi ? s1 : d1[1];
    const float dk = hi ? s2 : d1[2];

    // Reference semantics: strictly >0 lower bound, <=dim-1 upper bound.
    const bool ok = (di > 0.0f) && (dj > 0.0f) && (dk > 0.0f) &&
                    (di <= (float)(DD - 1)) && (dj <= (float)(HH - 1)) &&
                    (dk <= (float)(WW - 1));

    const float fi = fminf(fmaxf(floorf(di), 0.0f), (float)(DD - 1));
    const float fj = fminf(fmaxf(floorf(dj), 0.0f), (float)(HH - 1));
    const float fk = fminf(fmaxf(floorf(dk), 0.0f), (float)(WW - 1));
    const float wi = di - fi, wj = dj - fj, wk = dk - fk;

    const int i0 = (int)fi, j0 = (int)fj, q0 = (int)fk;
    const int i1 = min(i0 + 1, DD - 1);
    const int j1 = min(j0 + 1, HH - 1);
    const int q1 = min(q0 + 1, WW - 1);

    const int r00 = (i0 * HH + j0) * WW;
    const int r10 = (i1 * HH + j0) * WW;
    const int r01 = (i0 * HH + j1) * WW;
    const int r11 = (i1 * HH + j1) * WW;

    const float c000 = vol[r00 + q0], c100 = vol[r10 + q0];
    const float c010 = vol[r01 + q0], c110 = vol[r11 + q0];
    const float c001 = vol[r00 + q1], c101 = vol[r10 + q1];
    const float c011 = vol[r01 + q1], c111 = vol[r11 + q1];

    float r = ((c000 * (1.0f - wi) + c100 * wi) * (1.0f - wj) +
               (c010 * (1.0f - wi) + c110 * wi) * wj) * (1.0f - wk)
            + ((c001 * (1.0f - wi) + c101 * wi) * (1.0f - wj) +
               (c011 * (1.0f - wi) + c111 * wi) * wj) * wk;
    r = ok ? r : 0.0f;

    // Output is write-once: non-temporal store keeps the dead write stream
    // from displacing the (hot, fully L2-resident) source volume.
    __builtin_nontemporal_store(r, out + (ii * HH + jj) * WW + k0 + lane);
}

extern "C" void kernel_launch(void* const* d_in, const int* in_sizes, int n_in,
                              void* d_out, int out_size, void* d_ws, size_t ws_size,
                              hipStream_t stream) {
    const float* vol = (const float*)d_in[0];   // (1,1,192,224,192) f32
    const float* ang = (const float*)d_in[1];   // (1,3)
    const float* tr  = (const float*)d_in[2];   // (1,3)
    const float* sc  = (const float*)d_in[3];   // (1,3)
    const float* ref = (const float*)d_in[4];   // (4,4)
    const float* flo = (const float*)d_in[5];   // (4,4)
    float* T = (float*)d_ws;                    // 16 floats of scratch

    compose_T_kernel<<<1, 32, 0, stream>>>(ang, tr, sc, ref, flo, T);

    const int totalWaves = DD * HH * (WW / 32);       // 258048
    const int blocks     = totalWaves / (256 / 32);   // 32256
    resample_affine_kernel<<<blocks, 256, 0, stream>>>(vol, T, (float*)d_out);
}